// FEM_22574348108483
// MI455X (gfx1250) — compile-verified
//
#include <hip/hip_runtime.h>
#include <hip/hip_bf16.h>
#include <stdint.h>

// ---------------------------------------------------------------------------
// Fused 3-scale patch instance-norm attention for x:(16,64,256,256) f32.
// One workgroup (1024 threads = 32 wave32) per (b,c) channel image (64K f32).
// Stage channel global->LDS with CDNA5 async-to-LDS loads (NT hint, shared
// base + offset immediates), then keep the thread's 64 elements in VGPRs for
// all three scales (s = 64, 128, 256). Per-wave (sum, sumsq) partials in a
// 256B LDS scratch aggregate to any scale because every wave's elements lie
// inside a single 64x64 patch. Output written with non-temporal b128 stores.
// HBM-bound: 256MB in + 256MB out => ~22us floor at 23.3 TB/s.
// ---------------------------------------------------------------------------

#define LOG2E 1.44269504088896340736f
#define EPS_IN 1e-5f

typedef __attribute__((ext_vector_type(4))) float v4f;  // native clang vector

__device__ __forceinline__ float fast_sigmoid(float z) {
  // sigmoid(z) = 1 / (1 + exp(-z)); native v_exp_f32 / v_rcp_f32
  float e = __builtin_amdgcn_exp2f(-z * LOG2E);
  return __builtin_amdgcn_rcpf(1.0f + e);
}

__global__ __launch_bounds__(1024)
void FEM_patch_attn_kernel(const float* __restrict__ x, float* __restrict__ out) {
  extern __shared__ float lds[];                 // 65536 floats tile + scratch
  float*  tile = lds;                            // 256 KB staging tile
  float2* wsum = (float2*)(lds + 65536);         // 32 per-wave partials

  const int t    = threadIdx.x;
  const int wid  = t >> 5;
  const int lane = t & 31;
  const int p    = t >> 6;                       // 64x64 patch id, 0..15
  const int q    = t & 63;                       // thread within patch
  const int pr   = (p >> 2) * 64;                // patch row origin
  const int pc   = (p & 3) * 64;                 // patch col origin
  const int sr   = q >> 4;                       // 0..3  (sub-row)
  const int sc   = (q & 15) * 4;                 // 0..60 (float4 column)

  // chunk j (0..15) covers floats [e0 + 1024*j, +4): row pr+4j+sr, col pc+sc
  const int e0 = (pr + sr) * 256 + pc + sc;

  const size_t base = (size_t)blockIdx.x * 65536;
  const float* gsrc = x + base;
  float*       gdst = out + base;

  // ---- async bulk copy: global -> LDS (tracked by ASYNCcnt) ----
  // ISA: INST_OFFSET is added to BOTH the LDS dest address and the global
  // source address, so one base pair + offset immediates covers all chunks.
  {
    uint64_t ga = (uint64_t)(uintptr_t)(gsrc + e0);
    uint32_t la = (uint32_t)(uintptr_t)(tile + e0); // low 32b = LDS offset
    #pragma unroll
    for (int j = 0; j < 16; ++j) {
      asm volatile("global_load_async_to_lds_b128 %0, %1, off offset:%c2 th:TH_LOAD_NT"
                   :: "v"(la), "v"(ga), "n"(4096 * j) : "memory");
    }
    asm volatile("s_wait_asynccnt 0" ::: "memory");
  }

  // pull this thread's 64 elements into VGPRs (conflict-free ds_load_b128)
  v4f v[16];
  #pragma unroll
  for (int j = 0; j < 16; ++j)
    v[j] = *(const v4f*)(tile + e0 + 1024 * j);

  // ---- three fused scales: s = 64, 128, 256 ----
  #pragma unroll
  for (int scale = 0; scale < 3; ++scale) {
    // per-thread sum / sumsq over its 64 register-resident elements
    float s0 = 0.0f, s1 = 0.0f;
    #pragma unroll
    for (int j = 0; j < 16; ++j) {
      v4f d = v[j];
      s0 += (d.x + d.y) + (d.z + d.w);
      s1 = fmaf(d.x, d.x, s1); s1 = fmaf(d.y, d.y, s1);
      s1 = fmaf(d.z, d.z, s1); s1 = fmaf(d.w, d.w, s1);
    }
    // wave32 butterfly reduction
    #pragma unroll
    for (int m = 16; m >= 1; m >>= 1) {
      s0 += __shfl_xor(s0, m, 32);
      s1 += __shfl_xor(s1, m, 32);
    }
    if (lane == 0) wsum[wid] = make_float2(s0, s1);
    __syncthreads();

    // aggregate the wave partials belonging to this thread's patch-at-scale
    float S0 = 0.0f, S1 = 0.0f, invN;
    if (scale == 0) {                 // s = 64: waves 2p, 2p+1
      float2 a = wsum[2 * p], b = wsum[2 * p + 1];
      S0 = a.x + b.x; S1 = a.y + b.y;
      invN = 1.0f / 4096.0f;
    } else if (scale == 1) {          // s = 128: 4 nested 64-patches
      int bp = (p >> 3) * 8 + ((p >> 1) & 1) * 2;
      #pragma unroll
      for (int k = 0; k < 4; ++k) {
        int pp = bp + (k & 1) + (k >> 1) * 4;
        float2 a = wsum[2 * pp], b = wsum[2 * pp + 1];
        S0 += a.x + b.x; S1 += a.y + b.y;
      }
      invN = 1.0f / 16384.0f;
    } else {                          // s = 256: whole channel
      #pragma unroll
      for (int k = 0; k < 32; ++k) {
        float2 a = wsum[k];
        S0 += a.x; S1 += a.y;
      }
      invN = 1.0f / 65536.0f;
    }
    __syncthreads();                  // scratch consumed; reusable next scale

    float mean = S0 * invN;
    float var  = fmaf(S1, invN, -mean * mean);   // biased variance
    float rs   = rsqrtf(var + EPS_IN);

    // out = 0.5*out + 0.5*attn*out = out * 0.5 * (1 + sigmoid((out-m)*rs))
    #pragma unroll
    for (int j = 0; j < 16; ++j) {
      v4f d = v[j];
      d.x = 0.5f * d.x * (1.0f + fast_sigmoid((d.x - mean) * rs));
      d.y = 0.5f * d.y * (1.0f + fast_sigmoid((d.y - mean) * rs));
      d.z = 0.5f * d.z * (1.0f + fast_sigmoid((d.z - mean) * rs));
      d.w = 0.5f * d.w * (1.0f + fast_sigmoid((d.w - mean) * rs));
      v[j] = d;
    }
  }

  // coalesced non-temporal write-back from registers (streaming output)
  #pragma unroll
  for (int j = 0; j < 16; ++j)
    __builtin_nontemporal_store(v[j], (v4f*)(gdst + e0 + 1024 * j));
}

extern "C" void kernel_launch(void* const* d_in, const int* in_sizes, int n_in,
                              void* d_out, int out_size, void* d_ws, size_t ws_size,
                              hipStream_t stream) {
  const float* x = (const float*)d_in[0];
  float* out = (float*)d_out;
  const int nchan = in_sizes[0] >> 16;            // elements / 65536 = b*c
  const size_t shmem = 65536 * sizeof(float) + 32 * sizeof(float2); // 256.25 KB
  FEM_patch_attn_kernel<<<nchan, 1024, shmem, stream>>>(x, out);
}